// CLAHEEffect_41549513621543
// MI455X (gfx1250) — compile-verified
//
#include <hip/hip_runtime.h>

#define G 8
#define BINS 256
#define HSZ 4096
#define TS 512
#define NT 64   // G*G tiles

typedef float v2f __attribute__((ext_vector_type(2)));
typedef float v8f __attribute__((ext_vector_type(8)));

// -------------------- init: zero histograms, seed min/max --------------------
__global__ void init_ws_kernel(unsigned* __restrict__ ghist,
                               unsigned* __restrict__ gmin,
                               unsigned* __restrict__ gmax) {
    int i = blockIdx.x * blockDim.x + threadIdx.x;
    if (i < NT * BINS) ghist[i] = 0u;
    if (i < NT) { gmin[i] = 0x7F7FFFFFu; gmax[i] = 0u; }  // +FLT_MAX bits / 0
}

// -------------------- pass 1: per-tile histogram + min/max --------------------
// grid = 64 tiles * 16 chunks; block = 256 threads; each block covers 512x32 of one tile.
__global__ __launch_bounds__(256) void hist_kernel(const float* __restrict__ img,
                                                   unsigned* __restrict__ ghist,
                                                   unsigned* __restrict__ gmin,
                                                   unsigned* __restrict__ gmax) {
    __shared__ unsigned lhist[BINS];
    const int tile  = blockIdx.x >> 4;   // 0..63
    const int chunk = blockIdx.x & 15;   // 0..15
    const int t = threadIdx.x;
    lhist[t] = 0u;
    __syncthreads();

    const int ty = tile >> 3, tx = tile & 7;
    const int col  = (t & 127) << 2;          // 0..508 step 4
    const int row0 = chunk * 32 + (t >> 7);   // 2 rows in flight
    const size_t plane = (size_t)HSZ * HSZ;

    float vmin = 3.4e38f, vmax = 0.0f;        // lum >= 0
#pragma unroll 4
    for (int i = 0; i < 16; ++i) {
        const int gy = ty * TS + row0 + i * 2;
        const size_t off = (size_t)gy * HSZ + (size_t)(tx * TS + col);
        const float4 r = *(const float4*)(img + off);
        const float4 g = *(const float4*)(img + plane + off);
        const float4 b = *(const float4*)(img + 2 * plane + off);
        float l[4];
        l[0] = (r.x + g.x + b.x) * (1.0f / 3.0f);
        l[1] = (r.y + g.y + b.y) * (1.0f / 3.0f);
        l[2] = (r.z + g.z + b.z) * (1.0f / 3.0f);
        l[3] = (r.w + g.w + b.w) * (1.0f / 3.0f);
#pragma unroll
        for (int j = 0; j < 4; ++j) {
            int bin = (int)(l[j] * 256.0f);
            bin = bin < 0 ? 0 : (bin > 255 ? 255 : bin);
            atomicAdd(&lhist[bin], 1u);
            vmin = fminf(vmin, l[j]);
            vmax = fmaxf(vmax, l[j]);
        }
    }
    // wave32 min/max reduction, then 1 atomic per wave (nonneg float == uint order)
#pragma unroll
    for (int m = 1; m < 32; m <<= 1) {
        vmin = fminf(vmin, __shfl_xor(vmin, m, 32));
        vmax = fmaxf(vmax, __shfl_xor(vmax, m, 32));
    }
    if ((t & 31) == 0) {
        atomicMin(&gmin[tile], __float_as_uint(vmin));
        atomicMax(&gmax[tile], __float_as_uint(vmax));
    }
    __syncthreads();
    atomicAdd(&ghist[tile * BINS + t], lhist[t]);
}

// -------------------- pass 2: clip/redistribute + WMMA cumsum --------------------
// cdf(64x256) = histClipped(64x256) x U(256x256), U[k][j] = (k<=j).
// One block, 1024 threads = 32 waves; each wave computes two 16x16 D-tiles with
// V_WMMA_F32_16X16X4_F32. Lower-triangle K blocks use a constant all-ones B; only
// the diagonal block computes the triangular indicator. Loop bounds are forced
// into SGPRs (readfirstlane) so EXEC is never masked around the WMMAs.
__global__ __launch_bounds__(1024) void cdf_kernel(const unsigned* __restrict__ ghist,
                                                   const unsigned* __restrict__ gmin,
                                                   const unsigned* __restrict__ gmax,
                                                   const float* __restrict__ alphas,
                                                   float* __restrict__ cdf,
                                                   float* __restrict__ ptmin,
                                                   float* __restrict__ pinvr,
                                                   float* __restrict__ palpha,
                                                   float* __restrict__ pinvtot,
                                                   unsigned* __restrict__ pvalid) {
    extern __shared__ float histF[];   // 64*256 f32 = 64 KB
    const int t = threadIdx.x;

    // per-tile params (independent of WMMA phase)
    if (t < NT) {
        const float tmin = __uint_as_float(gmin[t]);
        const float tmax = __uint_as_float(gmax[t]);
        const int valid = tmin < tmax;
        ptmin[t]  = tmin;
        pinvr[t]  = 1.0f / (valid ? (tmax - tmin) : 1.0f);
        palpha[t] = 0.5f + 0.5f * alphas[t];
        pvalid[t] = (unsigned)valid;
    }

    // clip limit: max(1, int(4.0 * 512*512 / 256)) = 4096
    const float cl = 4096.0f;
    const int tile = t >> 4;   // 16 threads per tile
    const int sub  = t & 15;
    float h[16];
    float part = 0.0f;
#pragma unroll
    for (int j = 0; j < 16; ++j) {
        h[j] = (float)ghist[tile * BINS + sub * 16 + j];
        part += fmaxf(h[j] - cl, 0.0f);
    }
    // reduce excess across the 16 lanes owning this tile (xor stays in 16-lane half)
#pragma unroll
    for (int m = 1; m < 16; m <<= 1) part += __shfl_xor(part, m, 32);
    const float add = part * (1.0f / 256.0f);
#pragma unroll
    for (int j = 0; j < 16; ++j)
        histF[tile * BINS + sub * 16 + j] = fminf(h[j], cl) + add;
    __syncthreads();

    // WMMA phase: wave w -> output tiles w and w+32 (linear ot = ib*16+jb)
    const int w     = t >> 5;
    const int lane  = t & 31;
    const int lhalf = lane >> 4;   // 0 | 1
    const int l15   = lane & 15;
    v2f bones; bones.x = 1.0f; bones.y = 1.0f;
    for (int rep = 0; rep < 2; ++rep) {
        const int ot = w + rep * 32;
        const int ib = __builtin_amdgcn_readfirstlane(ot >> 4);   // tile-row block
        const int jb = __builtin_amdgcn_readfirstlane(ot & 15);   // bin-col block
        v8f c = {0.f, 0.f, 0.f, 0.f, 0.f, 0.f, 0.f, 0.f};
        const float* arow = &histF[(ib * 16 + l15) * BINS];
        const int ncol  = jb * 16 + l15;  // this lane's global bin column
        const int kfull = jb * 16;        // SGPR: blocks fully below the diagonal

        // bulk: U block is all-ones
        for (int k0 = 0; k0 < kfull; k0 += 4) {
            const int ka = k0 + lhalf * 2;   // A element K = v + 2*(lane>=16)
            v2f a;
            a.x = arow[ka];
            a.y = arow[ka + 1];
            c = __builtin_amdgcn_wmma_f32_16x16x4_f32(false, a, false, bones,
                                                      (short)0, c, false, false);
        }
        // diagonal block: triangular indicator
#pragma unroll
        for (int d = 0; d < 4; ++d) {
            const int k0 = kfull + d * 4;
            const int ka = k0 + lhalf * 2;
            v2f a, b;
            a.x = arow[ka];
            a.y = arow[ka + 1];
            b.x = (ka     <= ncol) ? 1.0f : 0.0f;
            b.y = (ka + 1 <= ncol) ? 1.0f : 0.0f;
            c = __builtin_amdgcn_wmma_f32_16x16x4_f32(false, a, false, b,
                                                      (short)0, c, false, false);
        }
        // D layout: element v, lanes<16 -> M=v, lanes>=16 -> M=v+8; N = lane&15
#pragma unroll
        for (int v = 0; v < 8; ++v) {
            const int trow = ib * 16 + v + lhalf * 8;
            const int bcol = jb * 16 + l15;
            const float val = c[v];
            cdf[trow * BINS + bcol] = val;
            if (bcol == 255) pinvtot[trow] = 1.0f / val;  // cdf[:,255] = total
        }
    }
}

// -------------------- pass 3: apply --------------------
__device__ __forceinline__ float clamp01(float v) {
    return fminf(fmaxf(v, 0.0f), 1.0f);
}

__global__ __launch_bounds__(256) void apply_kernel(const float* __restrict__ img,
                                                    const float* __restrict__ cdf,
                                                    const float* __restrict__ ptmin,
                                                    const float* __restrict__ pinvr,
                                                    const float* __restrict__ palpha,
                                                    const float* __restrict__ pinvtot,
                                                    const unsigned* __restrict__ pvalid,
                                                    float* __restrict__ out) {
    const size_t q = (size_t)blockIdx.x * blockDim.x + threadIdx.x; // quad of 4 px
    const int y = (int)(q >> 10);
    const int x = (int)(q & 1023) << 2;
    const size_t plane = (size_t)HSZ * HSZ;
    const size_t off = (size_t)y * HSZ + x;

    const float4 r = *(const float4*)(img + off);
    const float4 g = *(const float4*)(img + plane + off);
    const float4 b = *(const float4*)(img + 2 * plane + off);

    const int tile = (y >> 9) * 8 + (x >> 9);  // 512-aligned quad -> single tile
    const float tmin = ptmin[tile];
    const float invr = pinvr[tile];
    const float al   = palpha[tile];
    const float invt = pinvtot[tile];
    const bool valid = pvalid[tile] != 0u;
    const float* crow = cdf + tile * BINS;

    float rr[4] = {r.x, r.y, r.z, r.w};
    float gg[4] = {g.x, g.y, g.z, g.w};
    float bb[4] = {b.x, b.y, b.z, b.w};
    float ro[4], go[4], bo[4];
#pragma unroll
    for (int j = 0; j < 4; ++j) {
        const float l = (rr[j] + gg[j] + bb[j]) * (1.0f / 3.0f);
        const float norm = (l - tmin) * invr;
        int idx = (int)(norm * 255.0f);
        idx = idx < 0 ? 0 : (idx > 255 ? 255 : idx);
        const float enh = crow[idx] * invt;
        const float e = al * enh + (1.0f - al) * l;
        const bool nz = l > 1e-5f;
        const float rl = nz ? 1.0f / l : 0.0f;
        const float fr = nz ? rr[j] * rl : 1.0f;
        const float fg = nz ? gg[j] * rl : 1.0f;
        const float fb = nz ? bb[j] * rl : 1.0f;
        ro[j] = clamp01(valid ? e * fr : rr[j]);
        go[j] = clamp01(valid ? e * fg : gg[j]);
        bo[j] = clamp01(valid ? e * fb : bb[j]);
    }
    *(float4*)(out + off)             = make_float4(ro[0], ro[1], ro[2], ro[3]);
    *(float4*)(out + plane + off)     = make_float4(go[0], go[1], go[2], go[3]);
    *(float4*)(out + 2 * plane + off) = make_float4(bo[0], bo[1], bo[2], bo[3]);
}

// -------------------- launch --------------------
extern "C" void kernel_launch(void* const* d_in, const int* in_sizes, int n_in,
                              void* d_out, int out_size, void* d_ws, size_t ws_size,
                              hipStream_t stream) {
    const float* img    = (const float*)d_in[0];
    const float* alphas = (const float*)d_in[1];
    float* out = (float*)d_out;

    char* ws = (char*)d_ws;
    unsigned* ghist  = (unsigned*)(ws);            // 64*256 u32 = 65536 B
    unsigned* gmin   = (unsigned*)(ws + 65536);    // 64 u32
    unsigned* gmax   = (unsigned*)(ws + 65792);    // 64 u32
    float*    cdf    = (float*)(ws + 66048);       // 64*256 f32 = 65536 B
    float*    ptmin  = (float*)(ws + 131584);      // 64 f32
    float*    pinvr  = (float*)(ws + 131840);
    float*    palpha = (float*)(ws + 132096);
    float*    pinvt  = (float*)(ws + 132352);
    unsigned* pvalid = (unsigned*)(ws + 132608);

    init_ws_kernel<<<(NT * BINS + 1023) / 1024, 1024, 0, stream>>>(ghist, gmin, gmax);
    hist_kernel<<<NT * 16, 256, 0, stream>>>(img, ghist, gmin, gmax);
    cdf_kernel<<<1, 1024, NT * BINS * sizeof(float), stream>>>(
        ghist, gmin, gmax, alphas, cdf, ptmin, pinvr, palpha, pinvt, pvalid);
    apply_kernel<<<(HSZ * (HSZ / 4)) / 256, 256, 0, stream>>>(
        img, cdf, ptmin, pinvr, palpha, pinvt, pvalid, out);
}